// MolCLREncoder_1494648619555
// MI455X (gfx1250) — compile-verified
//
#include <hip/hip_runtime.h>

typedef float v2f __attribute__((ext_vector_type(2)));
typedef float v8f __attribute__((ext_vector_type(8)));

// ---------------- elementwise / graph kernels ----------------

__global__ void k_embed(const int* __restrict__ atom, const int* __restrict__ chir,
                        const float* __restrict__ e1, const float* __restrict__ e2,
                        float* __restrict__ h, int N, int D) {
  int idx = blockIdx.x * blockDim.x + threadIdx.x;
  if (idx >= N * D) return;
  int n = idx / D, c = idx % D;
  h[idx] = e1[atom[n] * D + c] + e2[chir[n] * D + c];
}

__global__ void k_agg_init(const float* __restrict__ h, const float* __restrict__ ee1_sl,
                           const float* __restrict__ ee2_sl, float* __restrict__ agg,
                           int N, int D) {
  int idx = blockIdx.x * blockDim.x + threadIdx.x;
  if (idx >= N * D) return;
  int c = idx % D;
  agg[idx] = h[idx] + ee1_sl[c] + ee2_sl[c];
}

__global__ void k_scatter(const float* __restrict__ h, const int* __restrict__ row,
                          const int* __restrict__ col, const int* __restrict__ btype,
                          const int* __restrict__ bdir, const float* __restrict__ ee1_l,
                          const float* __restrict__ ee2_l, float* __restrict__ agg,
                          int E, int D) {
  int idx = blockIdx.x * blockDim.x + threadIdx.x;
  if (idx >= E * D) return;
  int e = idx / D, c = idx % D;
  float v = h[(long long)col[e] * D + c] + ee1_l[btype[e] * D + c] + ee2_l[bdir[e] * D + c];
  __hip_atomic_fetch_add(&agg[(long long)row[e] * D + c], v,
                         __ATOMIC_RELAXED, __HIP_MEMORY_SCOPE_AGENT);
}

// ---------------- WMMA f32 GEMM with async double-buffered staging ----------------
// C[M,Nc] = A[M,K] @ B[K,Nc] + bias, optional relu.

#define ASYNC_B128(ldsoff, gaddr)                                         \
  asm volatile("global_load_async_to_lds_b128 %0, %1, off"                \
               :: "v"(ldsoff), "v"(gaddr) : "memory")
#define ASYNC_B32(ldsoff, gaddr)                                          \
  asm volatile("global_load_async_to_lds_b32 %0, %1, off"                 \
               :: "v"(ldsoff), "v"(gaddr) : "memory")

template <bool RELU>
__global__ __launch_bounds__(256) void k_gemm_wmma(
    const float* __restrict__ A, const float* __restrict__ B,
    const float* __restrict__ bias, float* __restrict__ C,
    int M, int K, int Nc) {
  constexpr int BM = 64, BN = 128, BK = 16, PK = BK + 4;  // 80B rows: 16B-aligned, conflict-free
  __shared__ float As[2][BM][PK];   // [m][k]
  __shared__ float Bs[2][BN][PK];   // transposed: [n][k]

  const int tid = threadIdx.x;
  const int lane = tid & 31;
  const int w = tid >> 5;       // wave 0..7
  const int tr = w & 3;         // 16-row tile within BM
  const int tc = w >> 2;        // 64-col group within BN
  const int rowBase = blockIdx.x * BM;
  const int colBase = blockIdx.y * BN;
  const int lm = lane & 15;
  const int kh = (lane >> 4) << 1;  // per-half-wave K offset: 0 or 2

  v8f acc[4] = {};

  auto mma_tile = [&](int buf) {
#pragma unroll
    for (int kk = 0; kk < BK; kk += 4) {
      v2f a = *(const v2f*)&As[buf][tr * 16 + lm][kk + kh];
#pragma unroll
      for (int ct = 0; ct < 4; ++ct) {
        v2f b = *(const v2f*)&Bs[buf][tc * 64 + ct * 16 + lm][kk + kh];
        acc[ct] = __builtin_amdgcn_wmma_f32_16x16x4_f32(
            false, a, false, b, (short)0, acc[ct], false, false);
      }
    }
  };

  auto fill_guard = [&](int buf, int k0) {
#pragma unroll
    for (int i = 0; i < 4; ++i) {           // A tile: 64x16
      int t = tid + i * 256;
      int m = t >> 4, kk = t & 15;
      int gr = rowBase + m, gk = k0 + kk;
      As[buf][m][kk] = (gr < M && gk < K) ? A[(size_t)gr * K + gk] : 0.0f;
    }
#pragma unroll
    for (int i = 0; i < 8; ++i) {           // B tile: 16x128 (stored transposed)
      int t = tid + i * 256;
      int kk = t >> 7, n = t & 127;
      int gk = k0 + kk, gc = colBase + n;
      Bs[buf][n][kk] = (gk < K && gc < Nc) ? B[(size_t)gk * Nc + gc] : 0.0f;
    }
  };

  auto fill_async = [&](int buf, int k0) {
    // A: one b128 per thread (16B global chunk -> 16B-aligned LDS row chunk)
    {
      int m = tid >> 2, q = (tid & 3) * 4;
      unsigned loff = (unsigned)(unsigned long long)&As[buf][m][q];
      unsigned long long g =
          (unsigned long long)(const void*)(A + (size_t)(rowBase + m) * K + (k0 + q));
      ASYNC_B128(loff, g);
    }
    // B: eight b32 per thread (transposing copy: contiguous global, strided LDS)
#pragma unroll
    for (int i = 0; i < 8; ++i) {
      int t = tid + i * 256;
      int kk = t >> 7, n = t & 127;
      unsigned loff = (unsigned)(unsigned long long)&Bs[buf][n][kk];
      unsigned long long g =
          (unsigned long long)(const void*)(B + (size_t)(k0 + kk) * Nc + (colBase + n));
      ASYNC_B32(loff, g);
    }
  };

  const bool interior = (rowBase + BM <= M) && (colBase + BN <= Nc);
  const int Kfull = K & ~(BK - 1);
  const int T = Kfull / BK;

  if (interior && T > 0) {
    fill_async(0, 0);
    for (int t = 0; t < T; ++t) {
      if (t + 1 < T) {
        fill_async((t + 1) & 1, (t + 1) * BK);
        // 9 async ops just issued for t+1; tile t's 9 retire in order -> wait <= 9
        asm volatile("s_wait_asynccnt 0x9" ::: "memory");
      } else {
        asm volatile("s_wait_asynccnt 0x0" ::: "memory");
      }
      __syncthreads();          // tile t resident in LDS for all waves
      mma_tile(t & 1);
      __syncthreads();          // all waves done reading buf (t&1): safe to overwrite
    }
    if (Kfull < K) {            // zero-padded K tail, synchronous
      fill_guard(T & 1, Kfull);
      __syncthreads();
      mma_tile(T & 1);
    }
  } else {                      // edge blocks: fully guarded synchronous path
    for (int k0 = 0; k0 < K; k0 += BK) {
      fill_guard(0, k0);
      __syncthreads();
      mma_tile(0);
      __syncthreads();
    }
  }

  // epilogue: C VGPR j -> row (j | lane-half*8), col = lane&15
  const int rOff = (lane < 16) ? 0 : 8;
#pragma unroll
  for (int ct = 0; ct < 4; ++ct) {
    int c = colBase + tc * 64 + ct * 16 + lm;
    if (c >= Nc) continue;
    float bv = bias ? bias[c] : 0.0f;
#pragma unroll
    for (int j = 0; j < 8; ++j) {
      int r = rowBase + tr * 16 + j + rOff;
      if (r < M) {
        float v = acc[ct][j] + bv;
        if (RELU) v = fmaxf(v, 0.0f);
        C[(size_t)r * Nc + c] = v;
      }
    }
  }
}

// ---------------- BatchNorm (deterministic two-pass) ----------------

__global__ void k_bn_partial(const float* __restrict__ h2, float* __restrict__ partial,
                             int N, int D) {
  int b = blockIdx.x;
  int r0 = b * 256;
  int r1 = min(r0 + 256, N);
  for (int c = threadIdx.x; c < D; c += blockDim.x) {
    float s = 0.f, q = 0.f;
    for (int r = r0; r < r1; ++r) {
      float v = h2[(long long)r * D + c];
      s += v;
      q += v * v;
    }
    partial[(long long)b * 2 * D + c] = s;
    partial[(long long)b * 2 * D + D + c] = q;
  }
}

__global__ void k_bn_final(const float* __restrict__ partial, int nBlk, int N, int D,
                           float* __restrict__ mean, float* __restrict__ rstd, float eps) {
  int c = blockIdx.x * blockDim.x + threadIdx.x;
  if (c >= D) return;
  float s = 0.f, q = 0.f;
  for (int i = 0; i < nBlk; ++i) {
    s += partial[(long long)i * 2 * D + c];
    q += partial[(long long)i * 2 * D + D + c];
  }
  float m = s / (float)N;
  float v = q / (float)N - m * m;
  mean[c] = m;
  rstd[c] = rsqrtf(fmaxf(v, 0.0f) + eps);
}

__global__ void k_bn_apply(const float* __restrict__ h2, const float* __restrict__ mean,
                           const float* __restrict__ rstd, const float* __restrict__ gamma_l,
                           const float* __restrict__ beta_l, float* __restrict__ h,
                           int N, int D, int relu) {
  int idx = blockIdx.x * blockDim.x + threadIdx.x;
  if (idx >= N * D) return;
  int c = idx % D;
  float v = gamma_l[c] * (h2[idx] - mean[c]) * rstd[c] + beta_l[c];
  if (relu) v = fmaxf(v, 0.0f);
  h[idx] = v;
}

// ---------------- per-graph mean pool (batch is sorted -> binary search) ----------------

__global__ void k_pool(const float* __restrict__ h, const int* __restrict__ batch,
                       float* __restrict__ pooled, int N, int D) {
  int g = blockIdx.x;
  int lo = 0, hi = N;
  while (lo < hi) { int mid = (lo + hi) >> 1; if (batch[mid] < g) lo = mid + 1; else hi = mid; }
  int start = lo;
  hi = N;
  while (lo < hi) { int mid = (lo + hi) >> 1; if (batch[mid] < g + 1) lo = mid + 1; else hi = mid; }
  int end = lo;
  float inv = 1.0f / fmaxf((float)(end - start), 1.0f);
  for (int c = threadIdx.x; c < D; c += blockDim.x) {
    float s = 0.f;
    for (int r = start; r < end; ++r) s += h[(long long)r * D + c];
    pooled[(long long)g * D + c] = s * inv;
  }
}

// ---------------- launcher ----------------

static inline size_t align256(size_t x) { return (x + 255) & ~(size_t)255; }

extern "C" void kernel_launch(void* const* d_in, const int* in_sizes, int n_in,
                              void* d_out, int out_size, void* d_ws, size_t ws_size,
                              hipStream_t stream) {
  const int* atom   = (const int*)d_in[0];
  const int* chir   = (const int*)d_in[1];
  const int* eidx   = (const int*)d_in[2];
  const int* btype  = (const int*)d_in[3];
  const int* bdir   = (const int*)d_in[4];
  const int* batch  = (const int*)d_in[5];
  const float* x_emb1 = (const float*)d_in[6];
  const float* x_emb2 = (const float*)d_in[7];
  const float* ee1    = (const float*)d_in[8];
  const float* ee2    = (const float*)d_in[9];
  const float* W1     = (const float*)d_in[10];
  const float* b1     = (const float*)d_in[11];
  const float* W2     = (const float*)d_in[12];
  const float* b2     = (const float*)d_in[13];
  const float* gamma  = (const float*)d_in[14];
  const float* beta   = (const float*)d_in[15];
  const float* feat_W = (const float*)d_in[16];
  const float* feat_b = (const float*)d_in[17];
  const float* proj_W = (const float*)d_in[18];
  const float* proj_b = (const float*)d_in[19];

  const int N    = in_sizes[0];
  const int E    = in_sizes[3];
  const int D    = in_sizes[6] / 119;
  const int L    = in_sizes[8] / (5 * D);
  const int FEAT = in_sizes[17];
  const int PROJ = in_sizes[19];
  const int G    = out_size / PROJ;
  const int D2   = 2 * D;

  const int* row = eidx;
  const int* col = eidx + E;

  // workspace carve
  char* p = (char*)d_ws;
  float* h   = (float*)p; p += align256((size_t)N * D * 4);
  float* agg = (float*)p; p += align256((size_t)N * D * 4);   // doubles as h2
  float* hid = (float*)p; p += align256((size_t)N * D2 * 4);  // reused for pooled/feat later
  const int nBlkStats = (N + 255) / 256;
  float* partial = (float*)p; p += align256((size_t)nBlkStats * 2 * D * 4);
  float* mean = (float*)p; p += align256((size_t)D * 4);
  float* rstd = (float*)p; p += align256((size_t)D * 4);
  float* pooled = hid;
  float* feat   = hid + align256((size_t)G * D * 4) / 4;
  (void)ws_size; (void)n_in;

  const int totND = N * D;
  const int totED = E * D;

  k_embed<<<(totND + 255) / 256, 256, 0, stream>>>(atom, chir, x_emb1, x_emb2, h, N, D);

  for (int l = 0; l < L; ++l) {
    const float* ee1_l = ee1 + (size_t)l * 5 * D;
    const float* ee2_l = ee2 + (size_t)l * 3 * D;
    // agg = h + (ee1[l,4] + ee2[l,0]) self-loop term, then scatter-add edge messages
    k_agg_init<<<(totND + 255) / 256, 256, 0, stream>>>(h, ee1_l + 4 * D, ee2_l, agg, N, D);
    k_scatter<<<(totED + 255) / 256, 256, 0, stream>>>(h, row, col, btype, bdir,
                                                       ee1_l, ee2_l, agg, E, D);
    // hid = relu(agg @ W1 + b1)
    dim3 g1((N + 63) / 64, (D2 + 127) / 128);
    k_gemm_wmma<true><<<g1, 256, 0, stream>>>(agg, W1 + (size_t)l * D * D2,
                                              b1 + (size_t)l * D2, hid, N, D, D2);
    // h2(=agg) = hid @ W2 + b2
    dim3 g2((N + 63) / 64, (D + 127) / 128);
    k_gemm_wmma<false><<<g2, 256, 0, stream>>>(hid, W2 + (size_t)l * D2 * D,
                                               b2 + (size_t)l * D, agg, N, D2, D);
    // BatchNorm (+ relu except last layer)
    k_bn_partial<<<nBlkStats, 256, 0, stream>>>(agg, partial, N, D);
    k_bn_final<<<(D + 255) / 256, 256, 0, stream>>>(partial, nBlkStats, N, D, mean, rstd, 1e-5f);
    k_bn_apply<<<(totND + 255) / 256, 256, 0, stream>>>(agg, mean, rstd,
                                                        gamma + (size_t)l * D,
                                                        beta + (size_t)l * D,
                                                        h, N, D, (l < L - 1) ? 1 : 0);
  }

  // global mean pool + two projection GEMMs
  k_pool<<<G, 256, 0, stream>>>(h, batch, pooled, N, D);
  dim3 g3((G + 63) / 64, (FEAT + 127) / 128);
  k_gemm_wmma<false><<<g3, 256, 0, stream>>>(pooled, feat_W, feat_b, feat, G, D, FEAT);
  dim3 g4((G + 63) / 64, (PROJ + 127) / 128);
  k_gemm_wmma<false><<<g4, 256, 0, stream>>>(feat, proj_W, proj_b, (float*)d_out, G, FEAT, PROJ);
}